// _QuantumKernel_65481071405264
// MI455X (gfx1250) — compile-verified
//
#include <hip/hip_runtime.h>
#include <hip/hip_bf16.h>

typedef __attribute__((ext_vector_type(2))) float v2f;
typedef __attribute__((ext_vector_type(8))) float v8f;

#define NWIRES 16
#define GROUPS 4   // 4 wires per group -> 2^4 = 16-dim feature per group
#define NJ 4       // j-tiles per wave (rolled loop; keeps VGPR pressure low)

// ---------------------------------------------------------------------------
// Stage 1: per-row feature expansion.
// U[g][i][t] = prod_{b=0..3} ( (t>>b)&1 ? sin(x[i,4g+b]/2) : cos(x[i,4g+b]/2) )
// Layout: U[((g*n + i) * 16) + t], row-major 16-wide per (g,i).
// ---------------------------------------------------------------------------
__global__ void qk_features(const float* __restrict__ X, float* __restrict__ U, int n) {
    int i = blockIdx.x * blockDim.x + threadIdx.x;
    if (i >= n) return;
    float c[NWIRES], s[NWIRES];
#pragma unroll
    for (int k = 0; k < NWIRES; ++k) {
        float ang = 0.5f * X[(size_t)i * NWIRES + k];
        __sincosf(ang, &s[k], &c[k]);
    }
#pragma unroll
    for (int g = 0; g < GROUPS; ++g) {
        int k0 = 4 * g;
#pragma unroll
        for (int t = 0; t < 16; ++t) {
            float p = 1.0f;
#pragma unroll
            for (int b = 0; b < 4; ++b)
                p *= ((t >> b) & 1) ? s[k0 + b] : c[k0 + b];
            U[((size_t)g * n + i) * 16 + t] = p;
        }
    }
}

// ---------------------------------------------------------------------------
// Stage 2: kernel[i,j] = | prod_g (U_g V_g^T)[i,j] | via V_WMMA_F32_16X16X4_F32.
// One wave per 16-row strip; holds A operands (4 groups x 4 K-steps x v2f) in
// registers and sweeps NJ consecutive 16-col tiles. The jj loop is kept rolled
// so only one tile's accumulators + B operands are live (no scratch spills).
//
// A-operand layout (16x4 f32): lane L, VGPR v  <-  A[L%16][2*(L/16)+v]
// B-operand (4x16 = V^T)     : lane L, VGPR v  <-  V[L%16][2*(L/16)+v]  (same!)
// D (16x16 f32)              : VGPR r -> row r + 8*(L/16), col L%16
// ---------------------------------------------------------------------------
__global__ void qk_wmma(const float* __restrict__ U, const float* __restrict__ V,
                        float* __restrict__ out, int n, int m) {
    const int lane = threadIdx.x & 31;
    const int wave = threadIdx.x >> 5;
    const int gw   = blockIdx.x * (blockDim.x >> 5) + wave;

    const int jStrips = m / (16 * NJ);
    const int ti  = gw / jStrips;          // i-tile index (0 .. n/16-1)
    const int tj0 = (gw % jStrips) * NJ;   // first j-tile of this wave's strip

    const int row    = lane & 15;          // M (for A) / N (for B) within tile
    const int colsel = (lane >> 4) << 1;   // K sub-column pair: 0 or 2

    // Load and pin the A operands for this i-tile: 4 groups x 4 K-steps.
    v2f a[GROUPS][4];
#pragma unroll
    for (int g = 0; g < GROUPS; ++g) {
        const float* Ur = U + ((size_t)g * n + ti * 16 + row) * 16 + colsel;
#pragma unroll
        for (int st = 0; st < 4; ++st)
            a[g][st] = *(const v2f*)(Ur + 4 * st);
    }

#pragma unroll 1
    for (int jj = 0; jj < NJ; ++jj) {
        const int tj = tj0 + jj;

        // Fetch all 16 B operands for this tile up-front (one s_clause-able
        // burst), then run the 16-WMMA chain.
        v2f b[GROUPS][4];
#pragma unroll
        for (int g = 0; g < GROUPS; ++g) {
            const float* Vr = V + ((size_t)g * m + tj * 16 + row) * 16 + colsel;
#pragma unroll
            for (int st = 0; st < 4; ++st)
                b[g][st] = *(const v2f*)(Vr + 4 * st);
        }

        v8f acc[GROUPS];
#pragma unroll
        for (int g = 0; g < GROUPS; ++g) {
            v8f c = {};
#pragma unroll
            for (int st = 0; st < 4; ++st) {
                // 8 args: (neg_a, A, neg_b, B, c_mod, C, reuse_a, reuse_b)
                c = __builtin_amdgcn_wmma_f32_16x16x4_f32(
                        false, a[g][st], false, b[g][st], (short)0, c, false, false);
            }
            acc[g] = c;
        }

        // Combine the four group GEMM tiles elementwise, abs, and store.
        const int orow0 = ti * 16 + ((lane >> 4) << 3);
        const int ocol  = tj * 16 + row;
#pragma unroll
        for (int r = 0; r < 8; ++r) {
            float vv = acc[0][r] * acc[1][r] * acc[2][r] * acc[3][r];
            out[(size_t)(orow0 + r) * m + ocol] = __builtin_fabsf(vv);
        }
    }
}

// ---------------------------------------------------------------------------
extern "C" void kernel_launch(void* const* d_in, const int* in_sizes, int n_in,
                              void* d_out, int out_size, void* d_ws, size_t ws_size,
                              hipStream_t stream) {
    const float* x = (const float*)d_in[0];
    const float* y = (const float*)d_in[1];
    float* out = (float*)d_out;

    const int n = in_sizes[0] / NWIRES;   // 2048
    const int m = in_sizes[1] / NWIRES;   // 2048

    float* U = (float*)d_ws;                       // GROUPS * n * 16 floats
    float* V = U + (size_t)GROUPS * n * 16;        // GROUPS * m * 16 floats

    qk_features<<<(n + 127) / 128, 128, 0, stream>>>(x, U, n);
    qk_features<<<(m + 127) / 128, 128, 0, stream>>>(y, V, m);

    const int waves  = (n / 16) * (m / (16 * NJ)); // 128 * 32 = 4096 waves
    const int blocks = waves / 8;                  // 8 waves (256 thr) per block
    qk_wmma<<<blocks, 256, 0, stream>>>(U, V, out, n, m);
}